// EQCNN_equi_39728447488681
// MI455X (gfx1250) — compile-verified
//
#include <hip/hip_runtime.h>
#include <math.h>

// ---------------------------------------------------------------------------
// VN-DGCNN forward for MI455X (gfx1250, wave32).
// WMMA (V_WMMA_F32_16X16X4_F32) for the kNN Gram matrices and the final
// 341x169 linear; gather-bound edge-conv passes use LDS-staged edges + VALU.
// K-dims are compile-time so the WMMA K-loops fully unroll with branchless
// tail masking (clamped row + select-zero) instead of exec-mask branches.
// ---------------------------------------------------------------------------

typedef float v2f __attribute__((ext_vector_type(2)));
typedef float v8f __attribute__((ext_vector_type(8)));

#define EPSN 1e-6f
#define EPSV 1e-5f
#define NEG  0.2f

// ---------------------------------------------------------------------------
// (B,N,3) -> (B,3,N)
__global__ void transpose_in(const float* __restrict__ x, float* __restrict__ h0, int N) {
  int n = blockIdx.x * blockDim.x + threadIdx.x;
  int b = blockIdx.y;
  if (n >= N) return;
  for (int z = 0; z < 3; ++z)
    h0[((long)b * 3 + z) * N + n] = x[((long)b * N + n) * 3 + z];
}

// xx[b][n] = sum_d F[b][d][n]^2
template <int D>
__global__ void xx_kernel(const float* __restrict__ F, long bS,
                          float* __restrict__ xx, int N) {
  int n = blockIdx.x * blockDim.x + threadIdx.x;
  int b = blockIdx.y;
  if (n >= N) return;
  const float* Fb = F + (long)b * bS + n;
  float s = 0.f;
#pragma unroll 8
  for (int d = 0; d < D; ++d) { float v = Fb[(long)d * N]; s += v * v; }
  xx[(long)b * N + n] = s;
}

// ---------------------------------------------------------------------------
// pd[b][i][j] = 2*G[i][j] - xx[i] - xx[j], G = X^T X  (one wave = 16x16 tile)
// f32 WMMA: A 16x4 (2 VGPR/lane: lanes 0-15 K={0,1}, lanes 16-31 K={2,3}),
// B 4x16 mirrored, C/D 8 VGPR f32 per the CDNA5 layout tables.
template <int D>
__global__ __launch_bounds__(32) void gram_kernel(
    const float* __restrict__ F, const float* __restrict__ xx,
    float* __restrict__ pd, long bS, int N) {
  int b = blockIdx.z;
  const float* X = F + (long)b * bS;
  int lane = threadIdx.x & 31;
  int half = lane >> 4, l = lane & 15;
  int i0 = blockIdx.y * 16, j0 = blockIdx.x * 16;
  const float* Ai = X + i0 + l + (long)(half * 2) * N;  // rows {d0+2h, d0+2h+1}
  const float* Bj = X + j0 + l + (long)(half * 2) * N;
  v8f acc = {0.f, 0.f, 0.f, 0.f, 0.f, 0.f, 0.f, 0.f};

  constexpr int DFULL = (D / 4) * 4;
#pragma unroll
  for (int d0 = 0; d0 < DFULL; d0 += 4) {
    v2f a, bb;
    a.x  = Ai[(long)d0 * N];
    a.y  = Ai[(long)(d0 + 1) * N];
    bb.x = Bj[(long)d0 * N];
    bb.y = Bj[(long)(d0 + 1) * N];
    acc = __builtin_amdgcn_wmma_f32_16x16x4_f32(false, a, false, bb,
                                                (short)0, acc, false, false);
  }
  if constexpr (DFULL < D) {
    // tail chunk: clamp rows in-bounds, zero contributions via select
    int da = DFULL + half * 2;
    int r0 = (da     < D) ? da     : (D - 1);
    int r1 = (da + 1 < D) ? da + 1 : (D - 1);
    float m0 = (da     < D) ? 1.f : 0.f;
    float m1 = (da + 1 < D) ? 1.f : 0.f;
    const float* Ax = X + i0 + l;
    const float* Bx = X + j0 + l;
    v2f a, bb;
    a.x  = Ax[(long)r0 * N] * m0;
    a.y  = Ax[(long)r1 * N] * m1;
    bb.x = Bx[(long)r0 * N] * m0;
    bb.y = Bx[(long)r1 * N] * m1;
    acc = __builtin_amdgcn_wmma_f32_16x16x4_f32(false, a, false, bb,
                                                (short)0, acc, false, false);
  }

  const float* xxb = xx + (long)b * N;
  float xj = xxb[j0 + l];
  float* pdb = pd + (long)b * N * N;
#pragma unroll
  for (int r = 0; r < 8; ++r) {
    int m = r + 8 * half;
    int i = i0 + m, j = j0 + l;
    pdb[(long)i * N + j] = 2.f * acc[r] - xxb[i] - xj;
  }
}

// ---------------------------------------------------------------------------
// top-k(20) per row: one wave per row; 32 candidates per lane in registers.
__global__ __launch_bounds__(32) void topk_kernel(
    const float* __restrict__ pd, int* __restrict__ idx, int N, int Kk) {
  int b = blockIdx.y, i = blockIdx.x;
  int lane = threadIdx.x;
  const float* row = pd + ((long)b * N + i) * N;
  const int T = 32;  // N == 1024
  float v[32];
#pragma unroll
  for (int t = 0; t < T; ++t) v[t] = row[t * 32 + lane];
  int* out = idx + ((long)b * N + i) * Kk;
  for (int s = 0; s < Kk; ++s) {
    float bv = -3.4e38f; int bt = 0;
#pragma unroll
    for (int t = 0; t < T; ++t) if (v[t] > bv) { bv = v[t]; bt = t; }
    int bj = bt * 32 + lane;
#pragma unroll
    for (int off = 16; off > 0; off >>= 1) {
      float ov = __shfl_xor(bv, off, 32);
      int   oj = __shfl_xor(bj, off, 32);
      if (ov > bv || (ov == bv && oj < bj)) { bv = ov; bj = oj; }
    }
    if (lane == 0) out[s] = bj;
    if ((bj & 31) == lane) v[bj >> 5] = -3.4e38f;
  }
}

// ---------------------------------------------------------------------------
// Stats pass: per-channel sum / sumsq of ||p|| over (B,N,k).
// One thread = one (b,n,k); its edge vector lives in an LDS slab.
template <int CIN, int COUT>
__global__ __launch_bounds__(32) void edge_stats(
    const float* __restrict__ F, long bS,
    const int* __restrict__ idx, const float* __restrict__ W,
    float* __restrict__ gsum, float* __restrict__ gsumsq, int N, int Kk) {
  extern __shared__ float smem[];
  constexpr int C2 = 2 * CIN;
  float* sSum  = smem;               // [COUT]
  float* sSq   = smem + COUT;        // [COUT]
  float* eslab = smem + 2 * COUT;    // [32 * C2*3]
  int tid = threadIdx.x;
  for (int o = tid; o < 2 * COUT; o += 32) smem[o] = 0.f;
  __syncthreads();

  long e = (long)blockIdx.x * 32 + tid;   // flat (b,n,k)
  long bn = e / Kk;
  int n = (int)(bn % N); int b = (int)(bn / N);
  int j = idx[e];
  const float* Fb = F + (long)b * bS;
  float* E = eslab + (long)tid * (C2 * 3);
#pragma unroll
  for (int c = 0; c < CIN; ++c)
#pragma unroll
    for (int z = 0; z < 3; ++z) {
      float fc = Fb[(long)(c * 3 + z) * N + n];
      float fn = Fb[(long)(c * 3 + z) * N + j];
      E[c * 3 + z]          = fn - fc;
      E[(CIN + c) * 3 + z]  = fc;
    }
  for (int o = 0; o < COUT; ++o) {
    const float* Wr = W + (long)o * C2;
    float px = 0.f, py = 0.f, pz = 0.f;
#pragma unroll 4
    for (int c = 0; c < C2; ++c) {
      float w = Wr[c];
      px += w * E[c * 3 + 0];
      py += w * E[c * 3 + 1];
      pz += w * E[c * 3 + 2];
    }
    float nrm = sqrtf(px * px + py * py + pz * pz) + EPSN;
    atomicAdd(&sSum[o], nrm);
    atomicAdd(&sSq[o], nrm * nrm);
  }
  __syncthreads();
  for (int o = tid; o < COUT; o += 32) {
    atomicAdd(&gsum[o], sSum[o]);
    atomicAdd(&gsumsq[o], sSq[o]);
  }
}

// ---------------------------------------------------------------------------
// Apply pass: recompute p,d, apply vector BN + VN-LeakyReLU, mean over k.
// One block = one (b,n); edges for all k staged in LDS; thread = out channel.
template <int CIN, int COUT>
__global__ __launch_bounds__(128) void edge_apply(
    const float* __restrict__ F, long bSin,
    const int* __restrict__ idx,
    const float* __restrict__ W, const float* __restrict__ Dm,
    const float* __restrict__ gamma, const float* __restrict__ beta,
    const float* __restrict__ gsum, const float* __restrict__ gsumsq,
    float cntInv, float* __restrict__ out, long bSout, int N, int Kk) {
  extern __shared__ float E[];  // [Kk * C2 * 3]
  constexpr int C2 = 2 * CIN;
  int n = blockIdx.x, b = blockIdx.y, tid = threadIdx.x;
  const float* Fb = F + (long)b * bSin;
  const int* id = idx + ((long)b * N + n) * Kk;
  int tot = Kk * CIN * 3;
  for (int t = tid; t < tot; t += 128) {
    int z = t % 3; int c = (t / 3) % CIN; int k = t / (3 * CIN);
    int j = id[k];
    float fc = Fb[(long)(c * 3 + z) * N + n];
    float fn = Fb[(long)(c * 3 + z) * N + j];
    E[(k * C2 + c) * 3 + z]        = fn - fc;
    E[(k * C2 + CIN + c) * 3 + z]  = fc;
  }
  __syncthreads();
  int o = tid;
  if (o >= COUT) return;
  float mu  = gsum[o] * cntInv;
  float var = gsumsq[o] * cntInv - mu * mu;
  float istd = rsqrtf(var + EPSV);
  float ga = gamma[o], be = beta[o];
  const float* Wr = W + (long)o * C2;
  const float* Dr = Dm + (long)o * C2;
  float ax = 0.f, ay = 0.f, az = 0.f;
  for (int k = 0; k < Kk; ++k) {
    const float* Ek = E + (long)k * C2 * 3;
    float px = 0.f, py = 0.f, pz = 0.f, dx = 0.f, dy = 0.f, dz = 0.f;
#pragma unroll 4
    for (int c = 0; c < C2; ++c) {
      float w = Wr[c], dd = Dr[c];
      float ex = Ek[c * 3], ey = Ek[c * 3 + 1], ez = Ek[c * 3 + 2];
      px += w * ex;  py += w * ey;  pz += w * ez;
      dx += dd * ex; dy += dd * ey; dz += dd * ez;
    }
    float nrm = sqrtf(px * px + py * py + pz * pz) + EPSN;
    float nbn = (nrm - mu) * istd * ga + be;
    float s = nbn / nrm;
    px *= s; py *= s; pz *= s;
    float dot = px * dx + py * dy + pz * dz;
    float qx = px, qy = py, qz = pz;
    if (dot < 0.f) {
      float dsq = dx * dx + dy * dy + dz * dz;
      float t = dot / (dsq + EPSN);
      qx = px - t * dx; qy = py - t * dy; qz = pz - t * dz;
    }
    ax += NEG * px + (1.f - NEG) * qx;
    ay += NEG * py + (1.f - NEG) * qy;
    az += NEG * pz + (1.f - NEG) * qz;
  }
  float invK = 1.f / (float)Kk;
  float* ob = out + (long)b * bSout;
  ob[(long)(o * 3 + 0) * N + n] = ax * invK;
  ob[(long)(o * 3 + 1) * N + n] = ay * invK;
  ob[(long)(o * 3 + 2) * N + n] = az * invK;
}

// ---------------------------------------------------------------------------
// Layer 5: p5[b] (M x 3N) = W5 (M x KC) * cat[b] (KC x 3N) via f32 WMMA.
template <int KC>
__global__ __launch_bounds__(32) void gemm5(
    const float* __restrict__ cat, const float* __restrict__ W5,
    float* __restrict__ p5, int M, int Ncol) {
  int b = blockIdx.z;
  const float* X = cat + (long)b * KC * Ncol;
  float* P = p5 + (long)b * M * Ncol;
  int lane = threadIdx.x & 31, half = lane >> 4, l = lane & 15;
  int o0 = blockIdx.y * 16, j0 = blockIdx.x * 16;
  int m = o0 + l;
  int mrow = (m < M) ? m : (M - 1);
  float amask = (m < M) ? 1.f : 0.f;
  const float* Arow = W5 + (long)mrow * KC + half * 2;  // cols {c0+2h, c0+2h+1}
  const float* Bj   = X + j0 + l + (long)(half * 2) * Ncol;
  v8f acc = {0.f, 0.f, 0.f, 0.f, 0.f, 0.f, 0.f, 0.f};

  constexpr int KFULL = (KC / 4) * 4;
#pragma unroll
  for (int c0 = 0; c0 < KFULL; c0 += 4) {
    v2f a, bb;
    a.x  = Arow[c0]     * amask;
    a.y  = Arow[c0 + 1] * amask;
    bb.x = Bj[(long)c0 * Ncol];
    bb.y = Bj[(long)(c0 + 1) * Ncol];
    acc = __builtin_amdgcn_wmma_f32_16x16x4_f32(false, a, false, bb,
                                                (short)0, acc, false, false);
  }
  if constexpr (KFULL < KC) {
    int ca = KFULL + half * 2;
    int r0 = (ca     < KC) ? ca     : (KC - 1);
    int r1 = (ca + 1 < KC) ? ca + 1 : (KC - 1);
    float m0 = (ca     < KC) ? amask : 0.f;
    float m1 = (ca + 1 < KC) ? amask : 0.f;
    float b0 = (ca     < KC) ? 1.f : 0.f;
    float b1 = (ca + 1 < KC) ? 1.f : 0.f;
    const float* Ar = W5 + (long)mrow * KC;
    const float* Bx = X + j0 + l;
    v2f a, bb;
    a.x  = Ar[r0] * m0;
    a.y  = Ar[r1] * m1;
    bb.x = Bx[(long)r0 * Ncol] * b0;
    bb.y = Bx[(long)r1 * Ncol] * b1;
    acc = __builtin_amdgcn_wmma_f32_16x16x4_f32(false, a, false, bb,
                                                (short)0, acc, false, false);
  }
#pragma unroll
  for (int r = 0; r < 8; ++r) {
    int o = o0 + r + 8 * half;
    if (o < M) P[(long)o * Ncol + j0 + l] = acc[r];
  }
}

// d5[b][col] = sum_c D5[c] * cat[b][c][col]   (Dout = 1)
template <int KC>
__global__ void dir5_kernel(const float* __restrict__ cat, const float* __restrict__ D5,
                            float* __restrict__ d5, int Ncol) {
  int t = blockIdx.x * blockDim.x + threadIdx.x;
  int b = blockIdx.y;
  if (t >= Ncol) return;
  const float* X = cat + (long)b * KC * Ncol + t;
  float acc = 0.f;
#pragma unroll 8
  for (int c = 0; c < KC; ++c) acc += D5[c] * X[(long)c * Ncol];
  d5[(long)b * Ncol + t] = acc;
}

__global__ void stats5_kernel(const float* __restrict__ p5,
                              float* __restrict__ gsum, float* __restrict__ gsumsq,
                              int M, int N) {
  __shared__ float s1[256], s2[256];
  int n = blockIdx.x * blockDim.x + threadIdx.x;
  int o = blockIdx.y, b = blockIdx.z;
  const float* P = p5 + ((long)b * M + o) * 3 * N;
  float px = P[n], py = P[N + n], pz = P[2 * N + n];
  float nrm = sqrtf(px * px + py * py + pz * pz) + EPSN;
  s1[threadIdx.x] = nrm; s2[threadIdx.x] = nrm * nrm;
  __syncthreads();
  for (int w = 128; w > 0; w >>= 1) {
    if ((int)threadIdx.x < w) {
      s1[threadIdx.x] += s1[threadIdx.x + w];
      s2[threadIdx.x] += s2[threadIdx.x + w];
    }
    __syncthreads();
  }
  if (threadIdx.x == 0) { atomicAdd(&gsum[o], s1[0]); atomicAdd(&gsumsq[o], s2[0]); }
}

__global__ void apply5_kernel(const float* __restrict__ p5, const float* __restrict__ d5,
                              const float* __restrict__ gamma, const float* __restrict__ beta,
                              const float* __restrict__ gsum, const float* __restrict__ gsumsq,
                              float cntInv, float* __restrict__ outp, int M, int N) {
  int n = blockIdx.x * blockDim.x + threadIdx.x;
  int o = blockIdx.y, b = blockIdx.z;
  const float* P  = p5 + ((long)b * M + o) * 3 * N;
  const float* Dv = d5 + (long)b * 3 * N;
  float px = P[n], py = P[N + n], pz = P[2 * N + n];
  float dx = Dv[n], dy = Dv[N + n], dz = Dv[2 * N + n];
  float mu  = gsum[o] * cntInv;
  float var = gsumsq[o] * cntInv - mu * mu;
  float istd = rsqrtf(var + EPSV);
  float nrm = sqrtf(px * px + py * py + pz * pz) + EPSN;
  float nbn = (nrm - mu) * istd * gamma[o] + beta[o];
  float s = nbn / nrm;
  px *= s; py *= s; pz *= s;
  float dot = px * dx + py * dy + pz * dz;
  float qx = px, qy = py, qz = pz;
  if (dot < 0.f) {
    float dsq = dx * dx + dy * dy + dz * dz;
    float t = dot / (dsq + EPSN);
    qx -= t * dx; qy -= t * dy; qz -= t * dz;
  }
  float* O = outp + ((long)b * M + o) * 3 * N;
  O[n]         = NEG * px + (1.f - NEG) * qx;
  O[N + n]     = NEG * py + (1.f - NEG) * qy;
  O[2 * N + n] = NEG * pz + (1.f - NEG) * qz;
}

// ---------------------------------------------------------------------------
// Host-side per-layer launcher (templated on channel sizes).
template <int CIN, int COUT>
static void launch_edge_layer(const float* F, long bS,
                              const float* w, const float* dmw,
                              const float* ga, const float* be,
                              const int* idx, float* gsum, float* gsq,
                              float* outSlice, long bSout,
                              int B, int N, int KNN, hipStream_t stream) {
  constexpr int C2 = 2 * CIN;
  constexpr int D  = 3 * CIN;
  size_t shm_s = (size_t)(32 * C2 * 3 + 2 * COUT) * sizeof(float);
  edge_stats<CIN, COUT><<<dim3((B * N * KNN) / 32), 32, shm_s, stream>>>(
      F, bS, idx, w, gsum, gsq, N, KNN);
  size_t shm_a = (size_t)(KNN * C2 * 3) * sizeof(float);
  float cntInv = 1.f / ((float)B * N * KNN);
  edge_apply<CIN, COUT><<<dim3(N, B), 128, shm_a, stream>>>(
      F, bS, idx, w, dmw, ga, be, gsum, gsq, cntInv, outSlice, bSout, N, KNN);
  (void)D;
}

template <int D>
static void launch_knn(const float* F, long bS, float* xx, float* pd, int* idx,
                       int B, int N, int KNN, hipStream_t stream) {
  xx_kernel<D><<<dim3(N / 256, B), 256, 0, stream>>>(F, bS, xx, N);
  gram_kernel<D><<<dim3(N / 16, N / 16, B), 32, 0, stream>>>(F, xx, pd, bS, N);
  topk_kernel<<<dim3(N, B), 32, 0, stream>>>(pd, idx, N, KNN);
}

// ---------------------------------------------------------------------------
extern "C" void kernel_launch(void* const* d_in, const int* in_sizes, int n_in,
                              void* d_out, int out_size, void* d_ws, size_t ws_size,
                              hipStream_t stream) {
  (void)in_sizes; (void)n_in; (void)out_size; (void)ws_size;
  const int B = 8, N = 1024, KNN = 20;

  const float* x = (const float*)d_in[0];
  const float *w[5], *dm[5], *ga[5], *be[5];
  for (int i = 0; i < 5; ++i) {
    w[i]  = (const float*)d_in[1 + 4 * i];
    dm[i] = (const float*)d_in[2 + 4 * i];
    ga[i] = (const float*)d_in[3 + 4 * i];
    be[i] = (const float*)d_in[4 + 4 * i];
  }

  float* ws = (float*)d_ws;
  size_t off = 0;
  auto alloc = [&](size_t nf) { float* p = ws + off; off += nf; return p; };
  float* h0  = alloc((size_t)B * 3 * N);           //  96 KB
  float* cat = alloc((size_t)B * 169 * 3 * N);     //  16.6 MB: x1|x2|x3|x4 slices
  float* pd  = alloc((size_t)B * N * N);           //  33.5 MB
  float* xx  = alloc((size_t)B * N);               //  32 KB
  int*   idx = (int*)alloc((size_t)B * N * KNN);   //  640 KB
  float* p5  = alloc((size_t)B * 341 * 3 * N);     //  33.5 MB
  float* d5v = alloc((size_t)B * 3 * N);           //  96 KB
  float* gsum = alloc(512);
  float* gsq  = alloc(512);

  transpose_in<<<dim3(N / 256, B), 256, 0, stream>>>(x, h0, N);

  const long catS = (long)169 * 3 * N;

  // ---- Layer 1: Cin=1 -> Cout=21, kNN on D=3 (h0), out -> cat[ch 0..21)
  launch_knn<3>(h0, (long)3 * N, xx, pd, idx, B, N, KNN, stream);
  hipMemsetAsync(gsum, 0, 512 * sizeof(float), stream);
  hipMemsetAsync(gsq,  0, 512 * sizeof(float), stream);
  launch_edge_layer<1, 21>(h0, (long)3 * N, w[0], dm[0], ga[0], be[0],
                           idx, gsum, gsq, cat + 0 * 3 * N, catS, B, N, KNN, stream);

  // ---- Layer 2: Cin=21 -> Cout=21, kNN on D=63 (x1), out -> cat[ch 21..42)
  launch_knn<63>(cat + 0 * 3 * N, catS, xx, pd, idx, B, N, KNN, stream);
  hipMemsetAsync(gsum, 0, 512 * sizeof(float), stream);
  hipMemsetAsync(gsq,  0, 512 * sizeof(float), stream);
  launch_edge_layer<21, 21>(cat + 0 * 3 * N, catS, w[1], dm[1], ga[1], be[1],
                            idx, gsum, gsq, cat + 21 * 3 * N, catS, B, N, KNN, stream);

  // ---- Layer 3: Cin=21 -> Cout=42, kNN on D=63 (x2), out -> cat[ch 42..84)
  launch_knn<63>(cat + 21 * 3 * N, catS, xx, pd, idx, B, N, KNN, stream);
  hipMemsetAsync(gsum, 0, 512 * sizeof(float), stream);
  hipMemsetAsync(gsq,  0, 512 * sizeof(float), stream);
  launch_edge_layer<21, 42>(cat + 21 * 3 * N, catS, w[2], dm[2], ga[2], be[2],
                            idx, gsum, gsq, cat + 42 * 3 * N, catS, B, N, KNN, stream);

  // ---- Layer 4: Cin=42 -> Cout=85, kNN on D=126 (x3), out -> cat[ch 84..169)
  launch_knn<126>(cat + 42 * 3 * N, catS, xx, pd, idx, B, N, KNN, stream);
  hipMemsetAsync(gsum, 0, 512 * sizeof(float), stream);
  hipMemsetAsync(gsq,  0, 512 * sizeof(float), stream);
  launch_edge_layer<42, 85>(cat + 42 * 3 * N, catS, w[3], dm[3], ga[3], be[3],
                            idx, gsum, gsq, cat + 84 * 3 * N, catS, B, N, KNN, stream);

  // ---- Layer 5: pointwise VN linear on cat (169 -> 341), dir has 1 channel.
  const int M = 341, Ncol = 3 * N;
  gemm5<169><<<dim3(Ncol / 16, (M + 15) / 16, B), 32, 0, stream>>>(cat, w[4], p5, M, Ncol);
  dir5_kernel<169><<<dim3((Ncol + 255) / 256, B), 256, 0, stream>>>(cat, dm[4], d5v, Ncol);
  hipMemsetAsync(gsum, 0, 512 * sizeof(float), stream);
  hipMemsetAsync(gsq,  0, 512 * sizeof(float), stream);
  stats5_kernel<<<dim3(N / 256, M, B), 256, 0, stream>>>(p5, gsum, gsq, M, N);
  float cnt5 = 1.f / ((float)B * N);
  apply5_kernel<<<dim3(N / 256, M, B), 256, 0, stream>>>(
      p5, d5v, ga[4], be[4], gsum, gsq, cnt5, (float*)d_out, M, N);
}